// SelfAttentionLayer_76974403879430
// MI455X (gfx1250) — compile-verified
//
#include <hip/hip_runtime.h>
#include <hip/hip_bf16.h>
#include <stdint.h>

// out = X @ Wv + bv.
// The reference's attention chain collapses: softmax rows sum to 1, so
// einsum('bqk,bql->bql', A, V) = rowsum(A) * V = V = X @ Wv + bv.
// Remaining work: one fp32 GEMM, M=8192 (B*S), N=1024, K=1024 (compute-bound,
// AI ~242 FLOP/B >> machine balance). fp32 WMMA (V_WMMA_F32_16X16X4_F32)
// keeps bit-level fidelity to the fp32 reference (bf16 would inject ~2e-3).
//
// Data movement: double-buffered LDS, software-pipelined K-tiles staged with
// GLOBAL_LOAD_ASYNC_TO_LDS_B128 (ASYNCcnt-tracked, no VGPR staging), with a
// register-staged pipeline as fallback if the builtin is absent.

typedef float v2f __attribute__((ext_vector_type(2)));
typedef float v4f __attribute__((ext_vector_type(4)));
typedef float v8f __attribute__((ext_vector_type(8)));
typedef int   v4i __attribute__((ext_vector_type(4)));

#define TILE_M 128
#define TILE_N 128
#define TILE_K 32
#define LDA 36    // padded row stride (floats) of As[TILE_M][LDA]; %4==0 -> 16B rows
#define LDB 136   // padded row stride (floats) of Bs[TILE_K][LDB]; 2*136%64==16 -> lane halves hit disjoint banks

#if __has_builtin(__builtin_amdgcn_global_load_async_to_lds_b128)
#define HAVE_ASYNC_LDS 1
#else
#define HAVE_ASYNC_LDS 0
#endif

__device__ __forceinline__ void wait_async_zero() {
#if __has_builtin(__builtin_amdgcn_s_wait_asynccnt)
  __builtin_amdgcn_s_wait_asynccnt(0);
#else
  asm volatile("s_wait_asynccnt 0x0" ::: "memory");
#endif
}

#if HAVE_ASYNC_LDS
// Builtin signature (from clang diagnostic): param0 = int4 addrspace(1)*
// ("__device__"), param1 = int4 addrspace(3)*, then imm offset + cpol.
typedef __attribute__((address_space(1))) v4i g_v4i;
typedef __attribute__((address_space(3))) v4i s_v4i;

// One 16B async copy per lane: global -> LDS, no VGPR staging, ASYNCcnt-tracked.
__device__ __forceinline__ void async_copy16(const float* g, float* l) {
  __builtin_amdgcn_global_load_async_to_lds_b128(
      (g_v4i*)(uintptr_t)g,                    // as(1) global source
      (s_v4i*)(uint32_t)(uintptr_t)l,          // low 32 bits of generic shared addr == LDS offset
      /*imm offset=*/0, /*cpol=*/0);
}
#endif

// 8 k-steps of K=4; 8 WMMAs per step; fragments per ISA 32-bit layouts:
// A 16x4: lane holds A[m][ka], A[m][ka+1] (m=lane&15, ka=kk+2*(lane>>4)).
// B 4x16: lane holds B[ka][n], B[ka+1][n]  (n=lane&15).
__device__ __forceinline__ void compute_tile(const float* __restrict__ as,
                                             const float* __restrict__ bs,
                                             v8f acc[2][4],
                                             int wr, int wc, int lm, int lh) {
#pragma unroll
  for (int kk = 0; kk < TILE_K; kk += 4) {
    const int ka = kk + 2 * lh;
    v2f a[2], b[4];
#pragma unroll
    for (int mi = 0; mi < 2; ++mi)
      a[mi] = *(const v2f*)&as[(wr * 32 + mi * 16 + lm) * LDA + ka];
#pragma unroll
    for (int ni = 0; ni < 4; ++ni) {
      const float* p = &bs[ka * LDB + wc * 64 + ni * 16 + lm];
      b[ni][0] = p[0];
      b[ni][1] = p[LDB];
    }
#pragma unroll
    for (int mi = 0; mi < 2; ++mi)
#pragma unroll
      for (int ni = 0; ni < 4; ++ni)
        acc[mi][ni] = __builtin_amdgcn_wmma_f32_16x16x4_f32(
            /*neg_a=*/false, a[mi], /*neg_b=*/false, b[ni],
            /*c_mod=*/(short)0, acc[mi][ni],
            /*reuse_a=*/false, /*reuse_b=*/false);
  }
}

__global__ __launch_bounds__(256)
void gemm_bias_wmma_f32(const float* __restrict__ X,
                        const float* __restrict__ W,
                        const float* __restrict__ bias,
                        float* __restrict__ Out,
                        int M, int N, int K) {
  __shared__ float As[2][TILE_M * LDA];   // 2 x 18.4 KB
  __shared__ float Bs[2][TILE_K * LDB];   // 2 x 17.4 KB

  const int tid  = threadIdx.x;
  const int lane = tid & 31;
  const int wave = tid >> 5;    // 8 waves
  const int wr   = wave >> 1;   // 0..3 : 32-row band
  const int wc   = wave & 1;    // 0..1 : 64-col band
  const int lm   = lane & 15;
  const int lh   = lane >> 4;

  const int m0 = blockIdx.y * TILE_M;
  const int n0 = blockIdx.x * TILE_N;

  v8f acc[2][4];
#pragma unroll
  for (int mi = 0; mi < 2; ++mi)
#pragma unroll
    for (int ni = 0; ni < 4; ++ni)
      acc[mi][ni] = (v8f){0.f, 0.f, 0.f, 0.f, 0.f, 0.f, 0.f, 0.f};

  float bn[4];
#pragma unroll
  for (int ni = 0; ni < 4; ++ni)
    bn[ni] = bias[n0 + wc * 64 + ni * 16 + lm];

  const int T = K / TILE_K;     // 32 K-tiles

#if HAVE_ASYNC_LDS
  // ---------------- async global->LDS pipelined path ----------------
  // Per tile: A = 128x32 f32, B = 32x128 f32; 4 b128 async copies per
  // matrix per thread.
#define LOAD_TILE_ASYNC(KT, BUF)                                            \
  do {                                                                      \
    _Pragma("unroll")                                                       \
    for (int i = 0; i < 4; ++i) {                                           \
      int id = tid + i * 256;                                               \
      int r = id >> 3, c = (id & 7) << 2;                                   \
      async_copy16(X + (size_t)(m0 + r) * K + (KT) + c,                     \
                   &As[BUF][r * LDA + c]);                                  \
    }                                                                       \
    _Pragma("unroll")                                                       \
    for (int i = 0; i < 4; ++i) {                                           \
      int id = tid + i * 256;                                               \
      int r = id >> 5, c = (id & 31) << 2;                                  \
      async_copy16(W + (size_t)((KT) + r) * N + n0 + c,                     \
                   &Bs[BUF][r * LDB + c]);                                  \
    }                                                                       \
  } while (0)

  LOAD_TILE_ASYNC(0, 0);
  wait_async_zero();
  __syncthreads();

  for (int t = 0; t < T; ++t) {
    const int cur = t & 1;
    if (t + 1 < T) LOAD_TILE_ASYNC((t + 1) * TILE_K, cur ^ 1);  // overlaps compute
    compute_tile(As[cur], Bs[cur], acc, wr, wc, lm, lh);
    if (t + 1 < T) {
      wait_async_zero();      // my async copies for tile t+1 have landed
      __syncthreads();        // everyone's have landed; buf cur is also free
    }
  }
#else
  // ---------------- register-staged pipelined fallback ----------------
  v4f ra[4], rb[4];
  int arr[4], arc[4], brr[4], brc[4];
#pragma unroll
  for (int i = 0; i < 4; ++i) {
    int id = tid + i * 256;
    arr[i] = id >> 3; arc[i] = (id & 7) << 2;
    brr[i] = id >> 5; brc[i] = (id & 31) << 2;
  }
#pragma unroll
  for (int i = 0; i < 4; ++i) {
    ra[i] = *(const v4f*)(X + (size_t)(m0 + arr[i]) * K + arc[i]);
    rb[i] = *(const v4f*)(W + (size_t)brr[i] * N + n0 + brc[i]);
  }
#pragma unroll
  for (int i = 0; i < 4; ++i) {
    *(v4f*)&As[0][arr[i] * LDA + arc[i]] = ra[i];
    *(v4f*)&Bs[0][brr[i] * LDB + brc[i]] = rb[i];
  }
  __syncthreads();

  for (int t = 0; t < T; ++t) {
    const int cur = t & 1;
    if (t + 1 < T) {
      const int kt = (t + 1) * TILE_K;
#pragma unroll
      for (int i = 0; i < 4; ++i) {
        ra[i] = *(const v4f*)(X + (size_t)(m0 + arr[i]) * K + kt + arc[i]);
        rb[i] = *(const v4f*)(W + (size_t)(kt + brr[i]) * N + n0 + brc[i]);
      }
    }
    compute_tile(As[cur], Bs[cur], acc, wr, wc, lm, lh);
    if (t + 1 < T) {
#pragma unroll
      for (int i = 0; i < 4; ++i) {
        *(v4f*)&As[cur ^ 1][arr[i] * LDA + arc[i]] = ra[i];
        *(v4f*)&Bs[cur ^ 1][brr[i] * LDB + brc[i]] = rb[i];
      }
      __syncthreads();
    }
  }
#endif

  // Epilogue: C/D 16x16 f32 layout: row = vgpr + 8*(lane>>4), col = lane&15.
#pragma unroll
  for (int mi = 0; mi < 2; ++mi) {
    const int rowBase = m0 + wr * 32 + mi * 16 + (lh << 3);
#pragma unroll
    for (int ni = 0; ni < 4; ++ni) {
      const int col = n0 + wc * 64 + ni * 16 + lm;
#pragma unroll
      for (int r = 0; r < 8; ++r)
        Out[(size_t)(rowBase + r) * N + col] = acc[mi][ni][r] + bn[ni];
    }
  }
}

extern "C" void kernel_launch(void* const* d_in, const int* in_sizes, int n_in,
                              void* d_out, int out_size, void* d_ws, size_t ws_size,
                              hipStream_t stream) {
  // setup_inputs order: X, Wq, bq, Wk, bk, Wv, bv
  const float* X  = (const float*)d_in[0];
  const float* Wv = (const float*)d_in[5];
  const float* bv = (const float*)d_in[6];
  float* out      = (float*)d_out;

  const int M = 4 * 2048;   // B*S
  const int N = 1024;       // D_OUT
  const int K = 1024;       // D_IN

  dim3 grid(N / TILE_N, M / TILE_M);   // 8 x 64 = 512 blocks
  gemm_bias_wmma_f32<<<grid, 256, 0, stream>>>(X, Wv, bv, out, M, N, K);
}